// linear_exp_76209899701080
// MI455X (gfx1250) — compile-verified
//
#include <hip/hip_runtime.h>
#include <stdint.h>

typedef __attribute__((ext_vector_type(2))) float v2f;
typedef __attribute__((ext_vector_type(8))) float v8f;

#define D_IN     4096
#define NPAD     16      // 10 real output cols padded to 16 for WMMA
#define WG_ROWS  128     // rows per workgroup (8 waves x 16)
#define BK       32      // K tile
#define NT       (D_IN / BK)   // 128 K-tiles
#define NBUF     4       // LDS pipeline buffers (3 tiles in flight + 1 compute)
#define XPITCH   36      // floats; 16B-aligned rows, conflict-free WMMA A reads
#define HPITCH   11

// ---- CDNA5 async global->LDS copies (ASYNCcnt, no VGPR round trip) ----
__device__ __forceinline__ void async_ld_b128(uint32_t lds_off, const void* g) {
    uint64_t ga = (uint64_t)g;
    asm volatile("global_load_async_to_lds_b128 %0, %1, off"
                 :: "v"(lds_off), "v"(ga) : "memory");
}
__device__ __forceinline__ void async_ld_b32(uint32_t lds_off, const void* g) {
    uint64_t ga = (uint64_t)g;
    asm volatile("global_load_async_to_lds_b32 %0, %1, off"
                 :: "v"(lds_off), "v"(ga) : "memory");
}
#define LDS_OFF(p) ((uint32_t)(uintptr_t)(p))   // low 32 bits of flat addr = LDS byte offset

__global__ __launch_bounds__(256) void fused_fc1_tt_fc2(
    const float* __restrict__ x,          // [16384, 4096]
    const float* __restrict__ fc1_w,      // [10, 4096]
    const float* __restrict__ fc1_b,      // [10]
    const float* __restrict__ core_first, // [1,2,2]  -> 4
    const float* __restrict__ cores_mid,  // [8,2,2,2]-> 64
    const float* __restrict__ core_last,  // [2,2,10] -> 40
    const float* __restrict__ fc2_w,      // [10,10]
    const float* __restrict__ fc2_b,      // [10]
    float* __restrict__ out)              // [16384, 10]
{
    __shared__ float xs[NBUF][WG_ROWS * XPITCH];  // 4 x 18432 B : x tiles
    __shared__ float ws[NBUF][BK * NPAD];         // 4 x  2048 B : weight tiles (transposed)
    __shared__ float hs[WG_ROWS * HPITCH];        //      5632 B : relu(fc1) rows

    const int tid     = threadIdx.x;
    const int lane    = tid & 31;
    const int laneLo  = lane & 15;
    const int laneHi  = lane >> 4;
    const int rowBase = blockIdx.x * WG_ROWS;
    // wave id as a proven-uniform SGPR value: scalar branches, no EXEC juggling
    const int waveU   = __builtin_amdgcn_readfirstlane(tid >> 5);
    const int waveRow = waveU * 16;

    // zero pad columns (n = 10..15) of every weight buffer once; asyncs never touch them
    for (int i = tid; i < NBUF * BK * NPAD; i += 256) {
        if ((i & (NPAD - 1)) >= 10) ((float*)ws)[i] = 0.0f;
    }

    // per-thread x-chunk coordinates: 128 rows x 8 b128-chunks = 1024 chunks, 4 per thread
    const int r0  = tid >> 3;              // row 0..31 (+32 per chunk set)
    const int cc0 = (tid & 7) * 4;         // col 0,4,..,28

    // issue all async copies for K-tile t (wave-uniform async counts:
    // waves 0..4 -> 4+2 = 6, waves 5..7 -> 4)
    auto issue = [&](int t) {
        const int buf = t & (NBUF - 1);
        const int k0  = t * BK;
#pragma unroll
        for (int i = 0; i < 4; ++i) {
            int r = r0 + i * 32;
            async_ld_b128(LDS_OFF(&xs[buf][r * XPITCH + cc0]),
                          x + (size_t)(rowBase + r) * D_IN + (k0 + cc0));
        }
        if (waveU < 5) {                   // scalar branch (waveU is SGPR)
            int n = waveU * 2;             // n = 0,2,4,6,8
            async_ld_b32(LDS_OFF(&ws[buf][lane * NPAD + n]),
                         fc1_w + (size_t)n * D_IN + (k0 + lane));
            async_ld_b32(LDS_OFF(&ws[buf][lane * NPAD + n + 1]),
                         fc1_w + (size_t)(n + 1) * D_IN + (k0 + lane));
        }
    };

    v8f acc = {0.f, 0.f, 0.f, 0.f, 0.f, 0.f, 0.f, 0.f};

    // prologue: 3 tiles in flight
    issue(0); issue(1); issue(2);

    for (int t = 0; t < NT; ++t) {
        // wait until tile t complete; tiles t+1, t+2 may remain outstanding
        // (async loads retire in order; per-wave issue counts are constant)
        if (waveU < 5) { asm volatile("s_wait_asynccnt 12" ::: "memory"); }
        else           { asm volatile("s_wait_asynccnt 8"  ::: "memory"); }
        __syncthreads();   // all waves' async writes for tile t landed; compute t-1 done
        if (t + 3 < NT) issue(t + 3);   // overwrites tile t-1's buffer: safe after barrier

        const int buf = t & (NBUF - 1);
        // ---- 8 x V_WMMA_F32_16X16X4_F32 over the BK=32 slab ----
        // A 16x4 fp32: lanes 0-15 -> M=laneLo, K=kk+{0,1}; lanes 16-31 -> K=kk+{2,3}
        // B 4x16 fp32: v0 = row K / K+2, v1 = row K+1 / K+3, N = laneLo
#pragma unroll
        for (int kk = 0; kk < BK; kk += 4) {
            const float* ap = &xs[buf][(waveRow + laneLo) * XPITCH + kk + 2 * laneHi];
            v2f a; a.x = ap[0]; a.y = ap[1];                 // ds_load_2addr_b64
            const float* bp = &ws[buf][(kk + 2 * laneHi) * NPAD + laneLo];
            v2f b; b.x = bp[0]; b.y = bp[NPAD];              // ds_load_2addr_b32
            acc = __builtin_amdgcn_wmma_f32_16x16x4_f32(
                false, a, false, b, (short)0, acc, false, false);
        }
    }

    __syncthreads();
    // ---- bias + relu; scatter C tile (VGPR r -> M=r / M=r+8) into hs ----
    if (laneLo < 10) {
        float bias = fc1_b[laneLo];
#pragma unroll
        for (int r = 0; r < 8; ++r) {
            int m = waveRow + laneHi * 8 + r;
            float v = acc[r] + bias;
            hs[m * HPITCH + laneLo] = fmaxf(v, 0.0f);
        }
    }
    __syncthreads();

    // ---- per-sample TT contraction + fc2 (threads 0..127 each own a sample) ----
    if (tid < WG_ROWS) {
        const float* h = &hs[tid * HPITCH];

        // core_first: [1,2,R] -> v = G[0,0,:] + h0 * G[0,1,:]
        float v0 = core_first[0] + h[0] * core_first[2];
        float v1 = core_first[1] + h[0] * core_first[3];

        // interior cores: [R,2,R], idx r*4 + m*2 + r'
#pragma unroll
        for (int k = 0; k < 8; ++k) {
            const float* M = cores_mid + k * 8;
            float hk = h[k + 1];
            float n0 = v0 * M[0] + v1 * M[4] + hk * (v0 * M[2] + v1 * M[6]);
            float n1 = v0 * M[1] + v1 * M[5] + hk * (v0 * M[3] + v1 * M[7]);
            v0 = n0; v1 = n1;
        }

        // core_last: [R,2,C], idx r*20 + m*10 + c
        float vc[10];
        float h9 = h[9];
#pragma unroll
        for (int c = 0; c < 10; ++c) {
            float t0 = v0 * core_last[c]      + v1 * core_last[20 + c];  // m=0
            float t1 = v0 * core_last[10 + c] + v1 * core_last[30 + c];  // m=1
            vc[c] = t0 + h9 * t1;
        }

        // fc2: out[j] = vc . fc2_w[j,:] + fc2_b[j]
        float* o = out + (size_t)(rowBase + tid) * 10;
#pragma unroll
        for (int j = 0; j < 10; ++j) {
            float s = fc2_b[j];
#pragma unroll
            for (int c = 0; c < 10; ++c) s += vc[c] * fc2_w[j * 10 + c];
            o[j] = s;
        }
    }
}

extern "C" void kernel_launch(void* const* d_in, const int* in_sizes, int n_in,
                              void* d_out, int out_size, void* d_ws, size_t ws_size,
                              hipStream_t stream) {
    const float* x          = (const float*)d_in[0];
    const float* fc1_w      = (const float*)d_in[1];
    const float* fc1_b      = (const float*)d_in[2];
    const float* core_first = (const float*)d_in[3];
    const float* cores_mid  = (const float*)d_in[4];
    const float* core_last  = (const float*)d_in[5];
    const float* fc2_w      = (const float*)d_in[6];
    const float* fc2_b      = (const float*)d_in[7];
    float* out = (float*)d_out;

    (void)in_sizes; (void)n_in; (void)out_size; (void)d_ws; (void)ws_size;

    const int B = 16384;
    fused_fc1_tt_fc2<<<B / WG_ROWS, 256, 0, stream>>>(
        x, fc1_w, fc1_b, core_first, cores_mid, core_last, fc2_w, fc2_b, out);
}